// VectorQuantizer_11957188952565
// MI455X (gfx1250) — compile-verified
//
#include <hip/hip_runtime.h>
#include <hip/hip_bf16.h>

// ---------------------------------------------------------------------------
// VQ-VAE vector quantizer for gfx1250 (MI455X).
//   z:         (16, 64, 64, 64) f32  (B, C, H, W)
//   embedding: (1024, 64)       f32
// Outputs concatenated in d_out (floats):
//   [0]                      loss
//   [1 .. 4194304]           z_q (B,C,H,W)
//   [4194305]                perplexity
//   [4194306 .. +67108863]   one-hot encodings (N, 1024)
//   [71303170 .. +65535]     argmin indices (as float)
// ---------------------------------------------------------------------------

#define N_POINTS 65536      // 16*64*64
#define NE       1024
#define EDIM     64
#define HWSZ     4096       // 64*64
#define BSTRIDE  262144     // 64*4096 (per-batch stride in z)

#define ZQ_OFF      1
#define PERP_OFF    4194305
#define ONEHOT_OFF  4194306
#define IDX_OFF     71303170

// d_ws float layout:
//   [0]            sum of squared diff accumulator
//   [1024..2047]   histogram (float counts)
//   [2048..3071]   codebook squared norms
//   [4096.. ]      indices (int), 65536 entries
#define WS_SUMSQ 0
#define WS_HIST  1024
#define WS_ENORM 2048
#define WS_IDX   4096

// LDS tile: 16 codebook rows x 64 floats, row stride padded to 68 so that a
// b64 fragment read hits all 64 LDS banks exactly once (4-bank lane stride).
#define EPITCH 68

typedef __attribute__((ext_vector_type(2))) float v2f;
typedef __attribute__((ext_vector_type(4))) float v4f;
typedef __attribute__((ext_vector_type(8))) float v8f;

// --------------------------------------------------------------------------
// Kernel 0: codebook norms + zero accumulators.
// --------------------------------------------------------------------------
__global__ __launch_bounds__(256) void vq_prep(const float* __restrict__ emb,
                                               float* __restrict__ ws) {
  int i = blockIdx.x * 256 + threadIdx.x;   // 0..1023
  if (i < NE) {
    const float* e = emb + (size_t)i * EDIM;
    float s = 0.f;
#pragma unroll 8
    for (int c = 0; c < EDIM; ++c) s += e[c] * e[c];
    ws[WS_ENORM + i] = s;
    ws[WS_HIST + i] = 0.f;
  }
  if (i == 0) ws[WS_SUMSQ] = 0.f;
}

// --------------------------------------------------------------------------
// Kernel 1: fused distance-GEMM + argmin via V_WMMA_F32_16X16X4_F32.
// Each wave owns 16 rows of zf (A-matrix resident in 32 VGPRs) and sweeps
// all 64 codebook column tiles. Codebook tiles are staged cooperatively
// into LDS (double-buffered) so the 8 waves share one coalesced global
// read per tile instead of 8 redundant gathers.
// score(n,e) = ||e||^2 - 2 * dot(z_n, e)   (||z||^2 is row-constant).
// --------------------------------------------------------------------------
__global__ __launch_bounds__(256) void vq_argmin(const float* __restrict__ z,
                                                 const float* __restrict__ emb,
                                                 const float* __restrict__ eNorm,
                                                 int* __restrict__ idx_out) {
  __shared__ float et[2][16 * EPITCH];   // double-buffered codebook tile

  const int tid   = threadIdx.x;
  const int lane  = tid & 31;
  const int wave  = tid >> 5;
  const int rowBase = blockIdx.x * 128 + wave * 16;
  const int nlane = lane & 15;          // N / M position within half-wave
  const int khalf = (lane >> 4) * 2;    // hi half-wave handles K+2, K+3

  // A-matrix fragments: 16x64 fp32 tile, ISA 16x4 layout per k-step.
  // zf[row][c] lives at z[b*BSTRIDE + c*HWSZ + hw].
  const int mrow = rowBase + nlane;
  const int b    = mrow >> 12;
  const int hw   = mrow & (HWSZ - 1);
  const float* zrow = z + (size_t)b * BSTRIDE + hw;

  v2f a[16];
#pragma unroll
  for (int ks = 0; ks < 16; ++ks) {
    const int k = ks * 4 + khalf;
    a[ks].x = zrow[(size_t)k * HWSZ];
    a[ks].y = zrow[(size_t)(k + 1) * HWSZ];
  }

  float minV[8];
  int   minI[8];
#pragma unroll
  for (int j = 0; j < 8; ++j) { minV[j] = 3.4e38f; minI[j] = 0; }

  // Stage tile 0: 256 threads x float4 = 4KB coalesced.
  {
    const float* src = emb + (size_t)tid * 4;
    const v4f v = *(const v4f*)src;
    et[0][(tid >> 4) * EPITCH + (tid & 15) * 4 + 0] = v.x;
    et[0][(tid >> 4) * EPITCH + (tid & 15) * 4 + 1] = v.y;
    et[0][(tid >> 4) * EPITCH + (tid & 15) * 4 + 2] = v.z;
    et[0][(tid >> 4) * EPITCH + (tid & 15) * 4 + 3] = v.w;
  }
  __syncthreads();

  for (int t = 0; t < 64; ++t) {
    const int cur = t & 1;

    // Prefetch next tile into the other buffer while this one is consumed.
    if (t < 63) {
      const float* src = emb + (size_t)(t + 1) * (16 * EDIM) + (size_t)tid * 4;
      const v4f v = *(const v4f*)src;
      float* dst = &et[cur ^ 1][(tid >> 4) * EPITCH + (tid & 15) * 4];
      *(v4f*)dst = v;
    }

    const int col = t * 16 + nlane;
    const float* erow = &et[cur][nlane * EPITCH + khalf];

    v8f acc = {};
#pragma unroll
    for (int ks = 0; ks < 16; ++ks) {
      const v2f bf = *(const v2f*)(erow + ks * 4);   // conflict-free ds_load_b64
      acc = __builtin_amdgcn_wmma_f32_16x16x4_f32(
          /*neg_a=*/false, a[ks], /*neg_b=*/false, bf,
          /*c_mod=*/(short)0, acc, /*reuse_a=*/false, /*reuse_b=*/false);
    }

    const float cn = eNorm[col];
#pragma unroll
    for (int j = 0; j < 8; ++j) {
      const float s = cn - 2.0f * acc[j];
      if (s < minV[j]) { minV[j] = s; minI[j] = col; }  // strict < keeps first
    }

    __syncthreads();   // one barrier/tile is enough for the 2-deep buffer
  }

  // C/D layout: VGPR j -> row M=j (lanes 0-15) and M=j+8 (lanes 16-31),
  // N = lane&15. Reduce across the 16 lanes of each half-wave.
#pragma unroll
  for (int j = 0; j < 8; ++j) {
    float v = minV[j];
    int   i = minI[j];
#pragma unroll
    for (int m = 8; m >= 1; m >>= 1) {
      const float ov = __shfl_xor(v, m, 32);
      const int   oi = __shfl_xor(i, m, 32);
      if (ov < v || (ov == v && oi < i)) { v = ov; i = oi; }
    }
    if (nlane == 0) {
      const int row = rowBase + j + (lane >> 4) * 8;
      idx_out[row] = i;
    }
  }
}

// --------------------------------------------------------------------------
// Kernel 2: one-hot encodings, nontemporal B64 stores (268 MB, > L2).
// 8 floats per thread; base is only 8-byte aligned so use float2 stores.
// --------------------------------------------------------------------------
__global__ __launch_bounds__(256) void vq_onehot(const int* __restrict__ idx,
                                                 float* __restrict__ oh) {
  const unsigned gid = blockIdx.x * 256u + threadIdx.x;  // 8,388,608 threads
  const int n  = gid >> 7;          // 128 threads per row
  const int e0 = (gid & 127) * 8;
  const int k  = idx[n];
  float* p = oh + (size_t)n * NE + e0;
#pragma unroll
  for (int q = 0; q < 4; ++q) {
    v2f v = {0.f, 0.f};
    const int base = e0 + q * 2;
    if (k == base)     v.x = 1.0f;
    if (k == base + 1) v.y = 1.0f;
    __builtin_nontemporal_store(v, (v2f*)(p + q * 2));
  }
}

// --------------------------------------------------------------------------
// Kernel 3: gather z_q (NCHW), squared-error reduction, histogram, indices.
// Forward loss value = (1 + BETA) * mean((z_q - z)^2).
// --------------------------------------------------------------------------
__global__ __launch_bounds__(256) void vq_zqloss(const float* __restrict__ z,
                                                 const float* __restrict__ emb,
                                                 const int* __restrict__ idx,
                                                 float* __restrict__ out,
                                                 float* __restrict__ ws) {
  const int t = blockIdx.x * 256 + threadIdx.x;  // 4,194,304 threads
  const int n = t >> 6;
  const int c = t & 63;
  const int b  = n >> 12;
  const int hw = n & (HWSZ - 1);
  const int k  = idx[n];

  const float zq = emb[(size_t)k * EDIM + c];
  const size_t zoff = (size_t)b * BSTRIDE + (size_t)c * HWSZ + hw;
  const float zv = z[zoff];
  const float d  = zq - zv;

  out[ZQ_OFF + zoff] = zq;
  if (c == 0) {
    atomicAdd(&ws[WS_HIST + k], 1.0f);
    out[IDX_OFF + n] = (float)k;
  }

  __shared__ float red[256];
  red[threadIdx.x] = d * d;
  __syncthreads();
#pragma unroll
  for (int s = 128; s > 0; s >>= 1) {
    if (threadIdx.x < s) red[threadIdx.x] += red[threadIdx.x + s];
    __syncthreads();
  }
  if (threadIdx.x == 0) atomicAdd(&ws[WS_SUMSQ], red[0]);
}

// --------------------------------------------------------------------------
// Kernel 4: finalize loss + perplexity.
// --------------------------------------------------------------------------
__global__ __launch_bounds__(256) void vq_final(const float* __restrict__ ws,
                                                float* __restrict__ out) {
  __shared__ float red[256];
  float acc = 0.f;
  for (int i = threadIdx.x; i < NE; i += 256) {
    const float em = ws[WS_HIST + i] * (1.0f / (float)N_POINTS);
    acc += em * __logf(em + 1e-10f);
  }
  red[threadIdx.x] = acc;
  __syncthreads();
#pragma unroll
  for (int s = 128; s > 0; s >>= 1) {
    if (threadIdx.x < s) red[threadIdx.x] += red[threadIdx.x + s];
    __syncthreads();
  }
  if (threadIdx.x == 0) {
    out[PERP_OFF] = __expf(-red[0]);
    out[0] = 1.25f * ws[WS_SUMSQ] / 4194304.0f;  // (1+BETA)*mean
  }
}

// --------------------------------------------------------------------------
extern "C" void kernel_launch(void* const* d_in, const int* in_sizes, int n_in,
                              void* d_out, int out_size, void* d_ws, size_t ws_size,
                              hipStream_t stream) {
  const float* z   = (const float*)d_in[0];
  const float* emb = (const float*)d_in[1];
  float* out = (float*)d_out;
  float* ws  = (float*)d_ws;
  int*   idx = (int*)(ws + WS_IDX);

  vq_prep<<<4, 256, 0, stream>>>(emb, ws);
  vq_argmin<<<N_POINTS / 128, 256, 0, stream>>>(z, emb, ws + WS_ENORM, idx);
  vq_onehot<<<(N_POINTS * 128) / 256, 256, 0, stream>>>(idx, out + ONEHOT_OFF);
  vq_zqloss<<<(N_POINTS * EDIM) / 256, 256, 0, stream>>>(z, emb, idx, out, ws);
  vq_final<<<1, 256, 0, stream>>>(ws, out);
}